// SimpleGraphBackbone_84301618086272
// MI455X (gfx1250) — compile-verified
//
#include <hip/hip_runtime.h>
#include <hip/hip_bf16.h>

#define N_NODES 100000
#define N_EDGES 1600000
#define DIM     128
#define NLAYERS 3
#define BN_EPS  1e-5f
#define LDA     132   // padded LDS row stride (floats): 132 % 64 = 4 banks/row -> conflict-free

typedef __attribute__((ext_vector_type(2))) float v2f;
typedef __attribute__((ext_vector_type(8))) float v8f;

// ---------------------------------------------------------------- utilities
__global__ void fill_f32(float* __restrict__ p, float v, int n) {
    int i = blockIdx.x * blockDim.x + threadIdx.x;
    if (i < n) p[i] = v;
}

__global__ void fill4_f32(float4* __restrict__ p, int n4) {
    int i = blockIdx.x * blockDim.x + threadIdx.x;
    if (i < n4) p[i] = make_float4(0.f, 0.f, 0.f, 0.f);
}

// ------------------------------------------------------------------ degrees
__global__ void deg_kernel(const int* __restrict__ dst, float* __restrict__ deg,
                           int n_edges) {
    int e = blockIdx.x * blockDim.x + threadIdx.x;
    if (e >= n_edges) return;
    int d = dst[e];
    if ((unsigned)d >= (unsigned)N_NODES) d = 0;   // defensive clamp
    atomicAdd(deg + d, 1.0f);
}

__global__ void invdeg_kernel(float* __restrict__ deg, int n) {
    int i = blockIdx.x * blockDim.x + threadIdx.x;
    if (i < n) deg[i] = 1.0f / fmaxf(deg[i], 1.0f);
}

// ---------------------------------------------------- mean-aggregate scatter
// One wave32 per edge; each lane moves one float4 (coalesced 512B per edge).
__global__ __launch_bounds__(256) void sage_scatter(
    const float* __restrict__ x, const int* __restrict__ src,
    const int* __restrict__ dst, float* __restrict__ agg, int n_edges) {
    int edge = blockIdx.x * (blockDim.x >> 5) + (threadIdx.x >> 5);
    int lane = threadIdx.x & 31;
    if (edge >= n_edges) return;
    int s = src[edge];
    int d = dst[edge];
    if ((unsigned)s >= (unsigned)N_NODES) s = 0;   // defensive clamp
    if ((unsigned)d >= (unsigned)N_NODES) d = 0;
    float4 v = *(const float4*)(x + (size_t)s * DIM + lane * 4);
    float* a = agg + (size_t)d * DIM + lane * 4;
    atomicAdd(a + 0, v.x);
    atomicAdd(a + 1, v.y);
    atomicAdd(a + 2, v.z);
    atomicAdd(a + 3, v.w);
}

// --------------------------------------------------------------- fused GEMM
// h = (agg * inv_deg) @ Wl^T + bl + x @ Wr^T, plus BN partial sums.
// Block: 16-node tile, 256 threads = 8 waves; wave w owns output channels
// [16w, 16w+16). 64 x v_wmma_f32_16x16x4_f32 per wave (f32 in, f32 acc).
__global__ __launch_bounds__(256) void sage_gemm(
    const float* __restrict__ agg, const float* __restrict__ xin,
    const float* __restrict__ inv_deg,
    const float* __restrict__ Wl, const float* __restrict__ Wr,
    const float* __restrict__ bias,
    float* __restrict__ h,
    float* __restrict__ bn_sum, float* __restrict__ bn_sumsq) {
    __shared__ float lds[2][16 * LDA];   // [0]=scaled agg tile, [1]=x tile

    const int node0 = blockIdx.x * 16;
    const int tid   = threadIdx.x;

    // cooperative tile load: 16 rows x 128 cols, 8 floats per thread
    {
        int row = tid >> 4;
        int col = (tid & 15) * 8;
        size_t g = (size_t)(node0 + row) * DIM + col;
        float idg = inv_deg[node0 + row];
        float4 a0 = *(const float4*)(agg + g);
        float4 a1 = *(const float4*)(agg + g + 4);
        float4 x0 = *(const float4*)(xin + g);
        float4 x1 = *(const float4*)(xin + g + 4);
        float* la = &lds[0][row * LDA + col];
        la[0] = a0.x * idg; la[1] = a0.y * idg; la[2] = a0.z * idg; la[3] = a0.w * idg;
        la[4] = a1.x * idg; la[5] = a1.y * idg; la[6] = a1.z * idg; la[7] = a1.w * idg;
        float* lx = &lds[1][row * LDA + col];
        lx[0] = x0.x; lx[1] = x0.y; lx[2] = x0.z; lx[3] = x0.w;
        lx[4] = x1.x; lx[5] = x1.y; lx[6] = x1.z; lx[7] = x1.w;
    }
    __syncthreads();

    const int lane  = tid & 31;
    const int o0    = (tid >> 5) * 16;        // this wave's output-channel tile
    const int mrow  = lane & 15;              // A: M index; B: N index (same value)
    const int khalf = (lane >> 4) << 1;       // lanes 16..31 hold K+2

    v8f c = {};
    #pragma unroll
    for (int pass = 0; pass < 2; ++pass) {
        const float* __restrict__ W = pass ? Wr : Wl;
        #pragma unroll 8
        for (int kk = 0; kk < 32; ++kk) {
            int kb = (kk << 2) + khalf;
            v2f a = *(const v2f*)&lds[pass][mrow * LDA + kb];              // A 16x4
            v2f b = *(const v2f*)(W + (size_t)(o0 + mrow) * DIM + kb);     // B 4x16 = W^T
            c = __builtin_amdgcn_wmma_f32_16x16x4_f32(
                    false, a, false, b, (short)0, c, false, false);
        }
    }

    // epilogue: bias, BN partial stats, store
    const float bo = bias[o0 + mrow];
    float s = 0.f, s2 = 0.f;
    float vals[8];
    #pragma unroll
    for (int v = 0; v < 8; ++v) {
        float t = c[v] + bo;
        vals[v] = t;
        s  += t;
        s2 += t * t;
    }
    const int rbase = node0 + ((lane >> 4) << 3);   // lanes 16..31 -> rows M+8
    #pragma unroll
    for (int v = 0; v < 8; ++v)
        h[(size_t)(rbase + v) * DIM + o0 + mrow] = vals[v];
    atomicAdd(&bn_sum[o0 + mrow], s);
    atomicAdd(&bn_sumsq[o0 + mrow], s2);
}

// ----------------------------------------------------------- BN fold + apply
__global__ void bn_finalize(const float* __restrict__ bn_sum,
                            const float* __restrict__ bn_sumsq,
                            const float* __restrict__ gamma,
                            const float* __restrict__ beta,
                            float* __restrict__ scale, float* __restrict__ shift) {
    int d = threadIdx.x;
    const float invN = 1.0f / (float)N_NODES;
    float mean = bn_sum[d] * invN;
    float var  = bn_sumsq[d] * invN - mean * mean;
    float sc   = gamma[d] * rsqrtf(var + BN_EPS);
    scale[d] = sc;
    shift[d] = beta[d] - mean * sc;
}

__global__ __launch_bounds__(256) void bn_apply(
    const float* __restrict__ h, const float* __restrict__ scale,
    const float* __restrict__ shift, const float* __restrict__ resid,
    float* __restrict__ out, int use_resid) {
    size_t i = ((size_t)blockIdx.x * blockDim.x + threadIdx.x) * 4;
    if (i >= (size_t)N_NODES * DIM) return;
    int col = (int)(i & (DIM - 1));
    float4 v  = *(const float4*)(h + i);
    float4 sc = *(const float4*)(scale + col);
    float4 sh = *(const float4*)(shift + col);
    v.x = fmaxf(v.x * sc.x + sh.x, 0.f);
    v.y = fmaxf(v.y * sc.y + sh.y, 0.f);
    v.z = fmaxf(v.z * sc.z + sh.z, 0.f);
    v.w = fmaxf(v.w * sc.w + sh.w, 0.f);
    if (use_resid) {
        float4 r = *(const float4*)(resid + i);
        v.x = r.x + 0.5f * v.x;
        v.y = r.y + 0.5f * v.y;
        v.z = r.z + 0.5f * v.z;
        v.w = r.w + 0.5f * v.w;
    }
    *(float4*)(out + i) = v;
}

// ------------------------------------------------------------------- driver
extern "C" void kernel_launch(void* const* d_in, const int* in_sizes, int n_in,
                              void* d_out, int out_size, void* d_ws, size_t ws_size,
                              hipStream_t stream) {
    const float* x_in  = (const float*)d_in[0];
    const int*   edges = (const int*)d_in[1];     // [2, N_EDGES]
    const float* Wl    = (const float*)d_in[2];   // [L, D, D]
    const float* bl    = (const float*)d_in[3];   // [L, D]
    const float* Wr    = (const float*)d_in[4];   // [L, D, D]
    const float* gamma = (const float*)d_in[5];   // [L, D]
    const float* beta  = (const float*)d_in[6];   // [L, D]
    float* out = (float*)d_out;

    const int* src = edges;
    const int* dst = edges + N_EDGES;

    const size_t ND   = (size_t)N_NODES * DIM;           // 12.8M floats
    const size_t NDB  = ND * sizeof(float);              // 51.2 MB
    char* ws = (char*)d_ws;
    float* agg     = (float*)(ws);
    float* hbuf0   = (float*)(ws + NDB);
    float* hbuf1   = (float*)(ws + 2 * NDB);
    float* inv_deg = (float*)(ws + 3 * NDB);
    size_t statoff = 3 * NDB + (((size_t)N_NODES * sizeof(float) + 255) & ~(size_t)255);
    float* bn_sum   = (float*)(ws + statoff);
    float* bn_sumsq = bn_sum + DIM;
    float* scale    = bn_sum + 2 * DIM;
    float* shift    = bn_sum + 3 * DIM;

    // ---- degrees (dst is the same every layer, as in the reference)
    fill_f32<<<(N_NODES + 255) / 256, 256, 0, stream>>>(inv_deg, 0.f, N_NODES);
    deg_kernel<<<(N_EDGES + 255) / 256, 256, 0, stream>>>(dst, inv_deg, N_EDGES);
    invdeg_kernel<<<(N_NODES + 255) / 256, 256, 0, stream>>>(inv_deg, N_NODES);

    const int scatter_blocks = (N_EDGES + 7) / 8;        // 1 wave32 per edge
    const int gemm_blocks    = N_NODES / 16;             // 100000 / 16 = 6250
    const int elem_blocks    = (int)((ND / 4 + 255) / 256);

    const float* xcur = x_in;
    float* hb[2] = {hbuf0, hbuf1};
    for (int l = 0; l < NLAYERS; ++l) {
        float* h = hb[l & 1];
        // zero agg + BN accumulators
        fill4_f32<<<elem_blocks, 256, 0, stream>>>((float4*)agg, (int)(ND / 4));
        fill_f32<<<1, 256, 0, stream>>>(bn_sum, 0.f, 2 * DIM);
        // mean aggregation
        sage_scatter<<<scatter_blocks, 256, 0, stream>>>(xcur, src, dst, agg, N_EDGES);
        // fused dual GEMM (+bias, +BN partial stats) with f32 WMMA
        sage_gemm<<<gemm_blocks, 256, 0, stream>>>(
            agg, xcur, inv_deg,
            Wl + (size_t)l * DIM * DIM, Wr + (size_t)l * DIM * DIM,
            bl + (size_t)l * DIM, h, bn_sum, bn_sumsq);
        // batch-norm fold + apply (+ReLU, +residual on middle layer)
        bn_finalize<<<1, DIM, 0, stream>>>(bn_sum, bn_sumsq,
                                           gamma + (size_t)l * DIM,
                                           beta + (size_t)l * DIM, scale, shift);
        float* o = (l == NLAYERS - 1) ? out : h;         // in-place BN except last
        bn_apply<<<elem_blocks, 256, 0, stream>>>(h, scale, shift, xcur, o,
                                                  (l == 1) ? 1 : 0);
        xcur = (l == NLAYERS - 1) ? out : h;
    }
}